// NOBlock_33079838114296
// MI455X (gfx1250) — compile-verified
//
#include <hip/hip_runtime.h>
#include <hip/hip_bf16.h>
#include <stdint.h>

// ---------------------------------------------------------------------------
// NOBlock fused kernel for MI455X (gfx1250, wave32).
//
// Math fusion: out[g] = (Xsum_g + Expand(core · Proj(Xsum_g))) * recipW_g,
// where Xsum_g = sum of the 4 tokens in quadtree group g (residual + group
// average are linear, mask only sets the divisor w). This cuts compute 4x and
// makes the kernel purely HBM-bound: read x once (256MB, non-temporal) +
// write out (64MB, non-temporal) => ~14us roofline @ 23.3 TB/s. core (1MB
// fp32) and factors stay L2-resident (regular-temporal) across all 512
// workgroups. Matrix ops use V_WMMA_F32_16X16X4_F32 (fp32 data).
// ---------------------------------------------------------------------------

typedef float v2f __attribute__((ext_vector_type(2)));
typedef float v4f __attribute__((ext_vector_type(4)));
typedef float v8f __attribute__((ext_vector_type(8)));

#define B_    4
#define N_    2048
#define M_    64          // grid cells per token (8x8)
#define C_    128         // channels
#define R_    8           // Tucker rank
#define NG    4           // quadtree groups handled per workgroup (16 tokens)
#define TPB   256         // 8 wave32 waves
#define NWAVE 8

// LDS plan (floats) -- total 161,792 B: two workgroups fit in 320KB WGP LDS.
#define SMEM_FLOATS (NG*8192 + NG*512 + NG*512 + 512 + 512 + 1024 + 1024 + 4*64 + NG*64)

__device__ __forceinline__ v8f wmma_f32_16x16x4(v2f a, v2f b, v8f c) {
    return __builtin_amdgcn_wmma_f32_16x16x4_f32(false, a, false, b, (short)0, c, false, false);
}

extern "C" __global__ __launch_bounds__(TPB)
void noblock_fused(const float* __restrict__ x,
                   const float* __restrict__ core,   // [512][512] = [ABC][DEF]
                   const float* __restrict__ of0,    // (8,8)
                   const float* __restrict__ of1,    // (8,8)
                   const float* __restrict__ of2,    // (128,8)
                   const float* __restrict__ if0,    // (8,8)
                   const float* __restrict__ if1,    // (8,8)
                   const float* __restrict__ if2,    // (128,8)
                   const unsigned char* __restrict__ mask, // (B,N,64) bool
                   float* __restrict__ out)          // (B, N/4, 64, 128)
{
    extern __shared__ float smem[];
    float* sX   = smem;                    // NG*8192  group-summed x tiles
    float* sZ   = sX   + NG*8192;          // NG*512   projected rank vectors
    float* sZ2  = sZ   + NG*512;           // NG*512   core GEMM result
    float* sA   = sZ2  + NG*512;           // 512      scratch
    float* sB   = sA   + 512;              // 512      scratch
    float* sIF2 = sB   + 512;              // 1024
    float* sOF2 = sIF2 + 1024;             // 1024
    float* sIF0 = sOF2 + 1024;             // 64
    float* sIF1 = sIF0 + 64;               // 64
    float* sOF0 = sIF1 + 64;               // 64
    float* sOF1 = sOF0 + 64;               // 64
    float* sW   = sOF1 + 64;               // NG*64    (w!=0)/(w+1e-10)

    const int tid  = threadIdx.x;
    const int lane = tid & 31;             // wave32
    const int wv   = tid >> 5;
    const int lo   = lane & 15;            // row/col within 16
    const int hi   = lane >> 4;            // half: K-pair select

    const int wg  = blockIdx.x;            // 0..511
    const int gpb = (N_/4)/NG;             // 128 group-blocks per batch
    const int b   = wg / gpb;
    const int g0  = (wg % gpb) * NG;       // first group index within batch b
    const int n0  = g0 * 4;                // first token within batch b

    // ---------------- phase 0: factors + mask weights ----------------------
    for (int i = tid; i < C_*R_; i += TPB) { sIF2[i] = if2[i]; sOF2[i] = of2[i]; }
    if (tid < 64) {
        sIF0[tid] = if0[tid]; sIF1[tid] = if1[tid];
        sOF0[tid] = of0[tid]; sOF1[tid] = of1[tid];
    }
    {   // one thread per (group, cell): w = #unmasked among the 4 tokens
        const int g = tid >> 6, m = tid & 63;
        const unsigned char* mp = mask + ((size_t)b*N_ + (size_t)(n0 + g*4))*M_ + m;
        float w = 0.f;
        #pragma unroll
        for (int j = 0; j < 4; ++j) w += mp[(size_t)j*M_] ? 0.f : 1.f;
        sW[tid] = (w > 0.f) ? (1.0f/(w + 1e-10f)) : 0.0f;
    }

    // ---------------- phase 1: Xsum per group (x read exactly once, NT) ----
    for (int g = 0; g < NG; ++g) {
        const size_t tokBase = ((size_t)b*N_ + (size_t)(n0 + g*4)) * (size_t)(M_*C_);
        v4f acc[8];
        #pragma unroll
        for (int u = 0; u < 8; ++u) acc[u] = (v4f){0.f,0.f,0.f,0.f};
        for (int j = 0; j < 4; ++j) {
            const v4f* xp = (const v4f*)(x + tokBase + (size_t)j*(M_*C_));
            #pragma unroll
            for (int u = 0; u < 8; ++u) {
                // non-temporal: read-once stream must not evict L2-resident core
                v4f v = __builtin_nontemporal_load(xp + tid + TPB*u);
                acc[u] += v;
            }
        }
        v4f* xs = (v4f*)(sX + g*8192);
        #pragma unroll
        for (int u = 0; u < 8; ++u) xs[tid + TPB*u] = acc[u];
    }
    __syncthreads();

    // ---------------- phase 2: Z_g = Proj(Xsum_g)  (rank-8, VALU) ----------
    for (int g = 0; g < NG; ++g) {
        const float* Xg = sX + g*8192;
        // 2a: T1[m=a*8+d][F] = sum_e Xsum[m][e] * in_f2[e][F]
        for (int o = tid; o < 512; o += TPB) {
            const int m = o >> 3, F = o & 7;
            float s = 0.f;
            for (int e = 0; e < C_; ++e) s += Xg[m*C_ + e] * sIF2[e*R_ + F];
            sA[o] = s;
        }
        __syncthreads();
        // 2b: U[a][E][F] = sum_d T1[a*8+d][F] * in_f1[d][E]
        for (int o = tid; o < 512; o += TPB) {
            const int a = o >> 6, E = (o >> 3) & 7, F = o & 7;
            float s = 0.f;
            #pragma unroll
            for (int d = 0; d < 8; ++d) s += sA[(a*8 + d)*8 + F] * sIF1[d*8 + E];
            sB[o] = s;                      // layout a*64 + E*8 + F
        }
        __syncthreads();
        // 2c: Z[D][E][F] = sum_a U[a][E][F] * in_f0[a][D]   (DEF packed)
        for (int o = tid; o < 512; o += TPB) {
            const int D = o >> 6, E = (o >> 3) & 7, F = o & 7;
            float s = 0.f;
            #pragma unroll
            for (int a = 0; a < 8; ++a) s += sB[a*64 + E*8 + F] * sIF0[a*8 + D];
            sZ[g*512 + o] = s;
        }
        __syncthreads();
    }

    // ---------------- phase 3: core GEMM via WMMA --------------------------
    // Z2[m][n] = sum_k Z[m][k] * core[n][k];  M=16 (rows>=NG masked to zero
    // with a per-lane multiply -- keeps EXEC all-ones / uniform control flow),
    // N=512 split 64/wave, K=512 in steps of 4. B = core^T read as b64 (RT,
    // L2-resident, reused by all 512 workgroups).
    {
        const int   nbase   = wv * 64;
        const float rowMask = (lo < NG) ? 1.f : 0.f;
        const int   arow    = (lo & (NG-1)) * 512;   // safe row for all lanes
        v8f acc[4];
        #pragma unroll
        for (int nt = 0; nt < 4; ++nt)
            #pragma unroll
            for (int r = 0; r < 8; ++r) acc[nt][r] = 0.f;

        for (int k0 = 0; k0 < 512; k0 += 4) {
            const int kk = k0 + hi*2;
            v2f a;                               // A: lane=row m, vgpr0/1 = kk,kk+1
            a[0] = sZ[arow + kk]     * rowMask;
            a[1] = sZ[arow + kk + 1] * rowMask;
            #pragma unroll
            for (int nt = 0; nt < 4; ++nt) {
                const float* bp = core + (size_t)(nbase + nt*16 + lo)*512 + kk;
                const v2f bb = *reinterpret_cast<const v2f*>(bp);  // 8B aligned
                acc[nt] = wmma_f32_16x16x4(a, bb, acc[nt]);
            }
        }
        // D rows 0..NG-1 live in VGPRs 0..3, lanes 0..15
        if (lane < 16) {
            #pragma unroll
            for (int nt = 0; nt < 4; ++nt)
                #pragma unroll
                for (int r = 0; r < NG; ++r)
                    sZ2[r*512 + nbase + nt*16 + lane] = acc[nt][r];
        }
    }
    __syncthreads();

    // ---------------- phase 4: expand + residual + weighted write ----------
    for (int g = 0; g < NG; ++g) {
        // 4a: t3[o][Bi][Ci] = sum_A Z2[A*64+Bi*8+Ci] * out_f0[o][A]
        for (int idx = tid; idx < 512; idx += TPB) {
            const int o = idx >> 6, Bi = (idx >> 3) & 7, Ci = idx & 7;
            float s = 0.f;
            #pragma unroll
            for (int A = 0; A < 8; ++A) s += sZ2[g*512 + A*64 + Bi*8 + Ci] * sOF0[o*8 + A];
            sA[idx] = s;
        }
        __syncthreads();
        // 4b: t4[m=o*8+p][Ci] = sum_Bi t3[o][Bi][Ci] * out_f1[p][Bi]
        for (int idx = tid; idx < 512; idx += TPB) {
            const int m = idx >> 3, Ci = idx & 7, o = m >> 3, p = m & 7;
            float s = 0.f;
            #pragma unroll
            for (int Bi = 0; Bi < 8; ++Bi) s += sA[o*64 + Bi*8 + Ci] * sOF1[p*8 + Bi];
            sB[idx] = s;                    // [64][8]
        }
        __syncthreads();
        // 4c: H[64][128] = t4[64][8] x out_f2^T[8][128] via WMMA (K=8),
        //     fused with residual + 1/w scale + non-temporal store.
        const size_t outBase = ((size_t)b*(N_/4) + (size_t)(g0 + g)) * (size_t)(M_*C_);
        for (int t = wv*4; t < wv*4 + 4; ++t) {     // 32 tiles / 8 waves
            const int mt = t >> 3, qt = t & 7;
            v8f acc;
            #pragma unroll
            for (int r = 0; r < 8; ++r) acc[r] = 0.f;
            #pragma unroll
            for (int k0 = 0; k0 < 8; k0 += 4) {
                const int kk = k0 + hi*2;
                v2f a, bv;
                a[0]  = sB[(mt*16 + lo)*R_ + kk];
                a[1]  = sB[(mt*16 + lo)*R_ + kk + 1];
                bv[0] = sOF2[(qt*16 + lo)*R_ + kk];
                bv[1] = sOF2[(qt*16 + lo)*R_ + kk + 1];
                acc = wmma_f32_16x16x4(a, bv, acc);
            }
            #pragma unroll
            for (int r = 0; r < 8; ++r) {           // D: vgpr r -> rows r / r+8
                const int m = mt*16 + r + hi*8;
                const int q = qt*16 + lo;
                const float val = (acc[r] + sX[g*8192 + m*C_ + q]) * sW[g*64 + m];
                __builtin_nontemporal_store(val, out + outBase + (size_t)m*C_ + q);
            }
        }
        __syncthreads();
    }
}

extern "C" void kernel_launch(void* const* d_in, const int* in_sizes, int n_in,
                              void* d_out, int out_size, void* d_ws, size_t ws_size,
                              hipStream_t stream) {
    (void)in_sizes; (void)n_in; (void)d_ws; (void)ws_size; (void)out_size;
    const float* x    = (const float*)d_in[0];
    const float* core = (const float*)d_in[1];
    const float* of0  = (const float*)d_in[2];
    const float* of1  = (const float*)d_in[3];
    const float* of2  = (const float*)d_in[4];
    const float* if0  = (const float*)d_in[5];
    const float* if1  = (const float*)d_in[6];
    const float* if2  = (const float*)d_in[7];
    const unsigned char* mask = (const unsigned char*)d_in[8];
    float* out = (float*)d_out;

    const dim3 grid(B_ * (N_/4) / NG);   // 512 workgroups, 16 tokens each
    const size_t shmem = (size_t)SMEM_FLOATS * sizeof(float);  // ~158 KB dynamic LDS
    noblock_fused<<<grid, TPB, shmem, stream>>>(x, core, of0, of1, of2,
                                                if0, if1, if2, mask, out);
}